// STGATEncoder_22471268893020
// MI455X (gfx1250) — compile-verified
//
#include <hip/hip_runtime.h>
#include <hip/hip_bf16.h>
#include <math.h>

// ---------------------------------------------------------------------------
// STGAT encoder for MI455X (gfx1250, wave32).
// GEMMs on the matrix pipe via V_WMMA_F32_16X16X4_F32 (fp32 end-to-end).
// Edge softmax/aggregation: wave-per-edge, b128 gathers (L2-resident),
// f32 global atomics for segment reductions.
// ---------------------------------------------------------------------------

typedef __attribute__((ext_vector_type(2))) float v2f;
typedef __attribute__((ext_vector_type(8))) float v8f;

#define HEADS 4
#define DDIM 128
#define NEG_SLOPE 0.2f

// Order-preserving float -> uint mapping for atomicMax-based segment-max.
__device__ __forceinline__ unsigned f2ord(float f) {
    unsigned u = __float_as_uint(f);
    return (u & 0x80000000u) ? ~u : (u | 0x80000000u);
}
__device__ __forceinline__ float ord2f(unsigned u) {
    return (u & 0x80000000u) ? __uint_as_float(u & 0x7fffffffu)
                             : __uint_as_float(~u);
}

// ---------------------------------------------------------------------------
// C[M x N] = act(A[M x K] @ W[K x N] + bias[N]);  one wave per 16x16 tile.
// fp32 WMMA, K consumed 4 at a time.  M, N multiples of 16; K multiple of 4.
// A operand layout (ISA 7.12.2, 32-bit A 16x4): lane%16 = row M,
//   lane/16 selects K pair {0,1} vs {2,3}; VGPR0=Keven, VGPR1=Kodd.
// B operand layout (32-bit B 4x16): lane%16 = col N, same K split.
// C/D layout: VGPR r -> row (lane/16)*8 + r, col lane%16.
// ---------------------------------------------------------------------------
__global__ void wmma_gemm_bias(const float* __restrict__ A,
                               const float* __restrict__ W,
                               const float* __restrict__ bias,
                               float* __restrict__ C,
                               int lda, int ldw, int ldc, int K, int do_relu) {
    const int mt   = blockIdx.x * 16;
    const int nt   = blockIdx.y * 16;
    const int lane = threadIdx.x;      // 0..31, one wave per block
    const int half = lane >> 4;        // 0 or 1
    const int l16  = lane & 15;

    const float* Arow = A + (size_t)(mt + l16) * lda;
    const float* Wcol = W + (nt + l16);

    v8f acc = {};
    for (int k0 = 0; k0 < K; k0 += 4) {
        const int ka = k0 + half * 2;
        v2f a, b;
        a.x = Arow[ka];
        a.y = Arow[ka + 1];
        b.x = Wcol[(size_t)ka * ldw];
        b.y = Wcol[(size_t)(ka + 1) * ldw];
        acc = __builtin_amdgcn_wmma_f32_16x16x4_f32(
            /*neg_a=*/false, a, /*neg_b=*/false, b,
            /*c_mod=*/(short)0, acc, /*reuse_a=*/false, /*reuse_b=*/false);
    }

    const float bv = bias[nt + l16];
    float* Crow = C + (size_t)(mt + half * 8) * ldc + nt + l16;
#pragma unroll
    for (int r = 0; r < 8; ++r) {
        float v = acc[r] + bv;
        if (do_relu) v = fmaxf(v, 0.0f);
        Crow[(size_t)r * ldc] = v;
    }
}

// ---------------------------------------------------------------------------
// Init: segment-max ordinals to ord(-inf), denominators to 0.
// ---------------------------------------------------------------------------
__global__ void init_seg_kernel(unsigned* __restrict__ mord,
                                float* __restrict__ denom, int n4) {
    int i = blockIdx.x * blockDim.x + threadIdx.x;
    if (i < n4) {
        mord[i]  = f2ord(-INFINITY);
        denom[i] = 0.0f;
    }
}

__global__ void zero_f32_kernel(float* __restrict__ p, int n) {
    int i = blockIdx.x * blockDim.x + threadIdx.x;
    if (i < n) p[i] = 0.0f;
}

// ---------------------------------------------------------------------------
// Pass A: per-edge attention logits + segment-max over dst.
// One wave per edge.  Lane owns channels 4*lane..4*lane+3 (float4 / b128
// gathers: one head row == 32 lanes x 16B).  Wave32 shuffle reduction.
// ---------------------------------------------------------------------------
__global__ void edge_logits_kernel(const float* __restrict__ xl,
                                   const float* __restrict__ xr,
                                   const int* __restrict__ ei,
                                   const float* __restrict__ att,
                                   float* __restrict__ logits,
                                   unsigned* __restrict__ mord,
                                   int E, int NE) {
    const int wid  = (int)((blockIdx.x * (size_t)blockDim.x + threadIdx.x) >> 5);
    const int lane = threadIdx.x & 31;
    if (wid >= NE) return;

    int s, d;
    if (wid < E) { s = ei[wid]; d = ei[E + wid]; }
    else         { s = d = wid - E; }

    const float4* xls = (const float4*)(xl + (size_t)s * (HEADS * DDIM)) + lane;
    const float4* xrd = (const float4*)(xr + (size_t)d * (HEADS * DDIM)) + lane;
    const float4* at4 = (const float4*)att + lane;

#pragma unroll
    for (int h = 0; h < HEADS; ++h) {
        const float4 a = xls[h * (DDIM / 4)];
        const float4 b = xrd[h * (DDIM / 4)];
        const float4 w = at4[h * (DDIM / 4)];
        float v0 = a.x + b.x, v1 = a.y + b.y, v2 = a.z + b.z, v3 = a.w + b.w;
        v0 = (v0 > 0.0f) ? v0 : v0 * NEG_SLOPE;
        v1 = (v1 > 0.0f) ? v1 : v1 * NEG_SLOPE;
        v2 = (v2 > 0.0f) ? v2 : v2 * NEG_SLOPE;
        v3 = (v3 > 0.0f) ? v3 : v3 * NEG_SLOPE;
        float part = v0 * w.x + v1 * w.y + v2 * w.z + v3 * w.w;
#pragma unroll
        for (int off = 16; off > 0; off >>= 1)
            part += __shfl_xor(part, off, 32);
        if (lane == 0) {
            logits[(size_t)wid * HEADS + h] = part;
            atomicMax(&mord[d * HEADS + h], f2ord(part));
        }
    }
}

// ---------------------------------------------------------------------------
// Pass B: p = exp(logit - m[dst]); denom[dst] += p.  Thread per (edge, head).
// ---------------------------------------------------------------------------
__global__ void edge_softmax_kernel(const int* __restrict__ ei,
                                    const unsigned* __restrict__ mord,
                                    float* __restrict__ logits_p,
                                    float* __restrict__ denom,
                                    int E, int NE) {
    int idx = blockIdx.x * blockDim.x + threadIdx.x;
    if (idx >= NE * HEADS) return;
    const int e = idx >> 2;
    const int h = idx & 3;
    const int d = (e < E) ? ei[E + e] : (e - E);
    float m = ord2f(mord[d * HEADS + h]);
    if (!__builtin_isfinite(m)) m = 0.0f;
    const float p = expf(logits_p[idx] - m);
    logits_p[idx] = p;
    atomicAdd(&denom[d * HEADS + h], p);
}

// ---------------------------------------------------------------------------
// Pass C: out[dst] += alpha * xl[src].  One wave per edge; float4 gather of
// the source row, 4 scalar f32 atomics per lane per head (no packed f32
// atomic exists; scatter stays global_atomic_add_f32).
// ---------------------------------------------------------------------------
__global__ void edge_aggregate_kernel(const float* __restrict__ xl,
                                      const int* __restrict__ ei,
                                      const float* __restrict__ p,
                                      const float* __restrict__ denom,
                                      float* __restrict__ outacc,
                                      int E, int NE) {
    const int wid  = (int)((blockIdx.x * (size_t)blockDim.x + threadIdx.x) >> 5);
    const int lane = threadIdx.x & 31;
    if (wid >= NE) return;

    int s, d;
    if (wid < E) { s = ei[wid]; d = ei[E + wid]; }
    else         { s = d = wid - E; }

    const float4* xls = (const float4*)(xl + (size_t)s * (HEADS * DDIM)) + lane;
    float* outd       = outacc + (size_t)d * (HEADS * DDIM) + lane * 4;

#pragma unroll
    for (int h = 0; h < HEADS; ++h) {
        const float dn    = denom[d * HEADS + h];
        const float alpha = p[(size_t)wid * HEADS + h] / fmaxf(dn, 1e-16f);
        const float4 a    = xls[h * (DDIM / 4)];
        float* o          = outd + h * DDIM;
        atomicAdd(o + 0, a.x * alpha);
        atomicAdd(o + 1, a.y * alpha);
        atomicAdd(o + 2, a.z * alpha);
        atomicAdd(o + 3, a.w * alpha);
    }
}

// ---------------------------------------------------------------------------
// gat[i][c] = mean over heads of out[i][h][c] + bias[c]
// One wave per node; lane owns 4 contiguous channels (b128 traffic).
// ---------------------------------------------------------------------------
__global__ void gat_mean_kernel(const float* __restrict__ outacc,
                                const float* __restrict__ bias,
                                float* __restrict__ gat, int Nn) {
    const int wid  = (int)((blockIdx.x * (size_t)blockDim.x + threadIdx.x) >> 5);
    const int lane = threadIdx.x & 31;
    if (wid >= Nn) return;
    const float4* o  = (const float4*)(outacc + (size_t)wid * (HEADS * DDIM)) + lane;
    const float4  bv = ((const float4*)bias)[lane];
    float4 s = o[0 * (DDIM / 4)];
#pragma unroll
    for (int h = 1; h < HEADS; ++h) {
        const float4 t = o[h * (DDIM / 4)];
        s.x += t.x; s.y += t.y; s.z += t.z; s.w += t.w;
    }
    float4 r;
    r.x = s.x * 0.25f + bv.x;
    r.y = s.y * 0.25f + bv.y;
    r.z = s.z * 0.25f + bv.z;
    r.w = s.w * 0.25f + bv.w;
    ((float4*)(gat + (size_t)wid * DDIM))[lane] = r;
}

// ---------------------------------------------------------------------------
// GRU cell on node 0.  Single block of 384 threads.
// ---------------------------------------------------------------------------
__global__ void gru_kernel(const float* __restrict__ gat,
                           const float* __restrict__ h0,
                           const float* __restrict__ W_ih,
                           const float* __restrict__ W_hh,
                           const float* __restrict__ b_ih,
                           const float* __restrict__ b_hh,
                           float* __restrict__ out) {
    __shared__ float gi[3 * DDIM];
    __shared__ float gh[3 * DDIM];
    const int t = threadIdx.x;
    if (t < 3 * DDIM) {
        float a = b_ih[t];
        float b = b_hh[t];
        const float4* wi = (const float4*)(W_ih + (size_t)t * DDIM);
        const float4* wh = (const float4*)(W_hh + (size_t)t * DDIM);
        const float4* g4 = (const float4*)gat;
        const float4* h4 = (const float4*)h0;
        for (int k = 0; k < DDIM / 4; ++k) {
            const float4 g = g4[k], w1 = wi[k];
            const float4 hh = h4[k], w2 = wh[k];
            a += g.x * w1.x + g.y * w1.y + g.z * w1.z + g.w * w1.w;
            b += hh.x * w2.x + hh.y * w2.y + hh.z * w2.z + hh.w * w2.w;
        }
        gi[t] = a;
        gh[t] = b;
    }
    __syncthreads();
    if (t < DDIM) {
        const float r  = 1.0f / (1.0f + expf(-(gi[t] + gh[t])));
        const float z  = 1.0f / (1.0f + expf(-(gi[DDIM + t] + gh[DDIM + t])));
        const float ng = tanhf(gi[2 * DDIM + t] + r * gh[2 * DDIM + t]);
        out[t] = (1.0f - z) * ng + z * h0[t];
    }
}

// ---------------------------------------------------------------------------
extern "C" void kernel_launch(void* const* d_in, const int* in_sizes, int n_in,
                              void* d_out, int out_size, void* d_ws, size_t ws_size,
                              hipStream_t stream) {
    const float* node_features = (const float*)d_in[0];   // N x 32
    const float* hidden_state  = (const float*)d_in[1];   // 1 x 128
    const float* W_enc         = (const float*)d_in[2];   // 32 x 128
    const float* b_enc         = (const float*)d_in[3];   // 128
    const float* W_l           = (const float*)d_in[4];   // 128 x 512
    const float* b_l           = (const float*)d_in[5];   // 512
    const float* W_r           = (const float*)d_in[6];   // 128 x 512
    const float* b_r           = (const float*)d_in[7];   // 512
    const float* att           = (const float*)d_in[8];   // 4 x 128
    const float* gat_bias      = (const float*)d_in[9];   // 128
    const float* W_ih          = (const float*)d_in[10];  // 384 x 128
    const float* W_hh          = (const float*)d_in[11];  // 384 x 128
    const float* b_ih          = (const float*)d_in[12];  // 384
    const float* b_hh          = (const float*)d_in[13];  // 384
    const int*   edge_index    = (const int*)d_in[14];    // 2 x E

    const int F_IN = 32;
    const int N  = in_sizes[0] / F_IN;       // 20000
    const int E  = in_sizes[14] / 2;         // 320000
    const int NE = E + N;                    // edges incl. self loops

    // Workspace layout (fp32 unless noted).  out_accum aliases xr (dead after
    // pass A); gat aliases h (dead after the projection GEMMs).
    char* ws = (char*)d_ws;
    size_t off = 0;
    float* h      = (float*)(ws + off); off += (size_t)N * DDIM * 4;          // N x 128
    float* xl     = (float*)(ws + off); off += (size_t)N * HEADS * DDIM * 4;  // N x 512
    float* xr     = (float*)(ws + off); off += (size_t)N * HEADS * DDIM * 4;  // N x 512
    float* logits = (float*)(ws + off); off += (size_t)NE * HEADS * 4;        // NE x 4
    unsigned* mord = (unsigned*)(ws + off); off += (size_t)N * HEADS * 4;     // N x 4
    float* denom  = (float*)(ws + off); off += (size_t)N * HEADS * 4;         // N x 4
    float* outacc = xr;   // reuse
    float* gat    = h;    // reuse

    // 1) h = relu(nf @ W_enc + b_enc)   [WMMA fp32, K=32]
    {
        dim3 grid(N / 16, DDIM / 16);
        wmma_gemm_bias<<<grid, 32, 0, stream>>>(node_features, W_enc, b_enc, h,
                                                F_IN, DDIM, DDIM, F_IN, 1);
    }
    // 2) xl = h @ W_l + b_l ; xr = h @ W_r + b_r   [WMMA fp32, K=128]
    {
        dim3 grid(N / 16, (HEADS * DDIM) / 16);
        wmma_gemm_bias<<<grid, 32, 0, stream>>>(h, W_l, b_l, xl,
                                                DDIM, HEADS * DDIM, HEADS * DDIM, DDIM, 0);
        wmma_gemm_bias<<<grid, 32, 0, stream>>>(h, W_r, b_r, xr,
                                                DDIM, HEADS * DDIM, HEADS * DDIM, DDIM, 0);
    }
    // 3) init segment-max / denom accumulators (every launch: graph-replay safe)
    {
        int n4 = N * HEADS;
        init_seg_kernel<<<(n4 + 255) / 256, 256, 0, stream>>>(mord, denom, n4);
    }
    // 4) Pass A: logits + segment-max
    {
        int blocks = (NE * 32 + 255) / 256;
        edge_logits_kernel<<<blocks, 256, 0, stream>>>(xl, xr, edge_index, att,
                                                       logits, mord, E, NE);
    }
    // 5) zero out_accum (aliases xr, which is now dead)
    {
        int n = N * HEADS * DDIM;
        zero_f32_kernel<<<(n + 255) / 256, 256, 0, stream>>>(outacc, n);
    }
    // 6) Pass B: exp + segment-sum of p
    {
        int n = NE * HEADS;
        edge_softmax_kernel<<<(n + 255) / 256, 256, 0, stream>>>(edge_index, mord,
                                                                 logits, denom, E, NE);
    }
    // 7) Pass C: weighted scatter-add
    {
        int blocks = (NE * 32 + 255) / 256;
        edge_aggregate_kernel<<<blocks, 256, 0, stream>>>(xl, edge_index, logits,
                                                          denom, outacc, E, NE);
    }
    // 8) head mean + bias (gat aliases h)
    {
        int n = N;
        gat_mean_kernel<<<(n * 32 + 255) / 256, 256, 0, stream>>>(outacc, gat_bias, gat, N);
    }
    // 9) GRU on node 0 -> d_out (128 floats)
    gru_kernel<<<1, 384, 0, stream>>>(gat, hidden_state, W_ih, W_hh, b_ih, b_hh,
                                      (float*)d_out);
}